// GCN_29755533427171
// MI455X (gfx1250) — compile-verified
//
#include <hip/hip_runtime.h>
#include <hip/hip_bf16.h>

typedef float v2f __attribute__((ext_vector_type(2)));
typedef float v8f __attribute__((ext_vector_type(8)));

// ---------------------------------------------------------------------------
// Degree / normalization kernels
// ---------------------------------------------------------------------------
__global__ void deg_init_kernel(float* __restrict__ deg, int n) {
    int i = blockIdx.x * blockDim.x + threadIdx.x;
    if (i < n) deg[i] = 1.0f;  // self-loop contributes 1 to every node
}

__global__ void deg_count_kernel(const long long* __restrict__ dst,
                                 float* __restrict__ deg, int E) {
    int e = blockIdx.x * blockDim.x + threadIdx.x;
    if (e < E) atomicAdd(&deg[(int)dst[e]], 1.0f);
}

__global__ void deg_rsqrt_kernel(float* __restrict__ deg, int n) {
    int i = blockIdx.x * blockDim.x + threadIdx.x;
    if (i < n) deg[i] = rsqrtf(deg[i]);  // deg >= 1 always (self-loops)
}

// ---------------------------------------------------------------------------
// Dense projection: Y[N,FOUT] = A[N,FIN] @ W[FIN,FOUT] via V_WMMA_F32_16X16X4_F32
// One wave (32 threads) computes a 16-row tile. N % 16 == 0 (100000 = 6250*16),
// so EXEC is all-1s as WMMA requires. fp32 end-to-end — matches reference.
//
// VGPR layouts per CDNA5 ISA 7.12.2:
//   A 16x4 f32 : lane half (>=16) selects K {0,1} vs {2,3}; v[0]=K even, v[1]=K odd
//   B 4x16 f32 : mirrored (row K striped across 16 lanes per half)
//   C/D 16x16  : 8 VGPRs, row M = r + 8*(lane>=16), col N = lane&15
// ---------------------------------------------------------------------------
template <int FIN, int FOUT>
__global__ __launch_bounds__(32) void wmma_gemm_kernel(const float* __restrict__ A,
                                                       const float* __restrict__ W,
                                                       float* __restrict__ Y) {
    const int lane = threadIdx.x;
    const int half = lane >> 4;   // 0: lanes 0-15, 1: lanes 16-31
    const int m    = lane & 15;
    const int row0 = blockIdx.x * 16;
    const float* arow = A + (size_t)(row0 + m) * FIN;

#pragma unroll
    for (int nt = 0; nt < FOUT / 16; ++nt) {
        v8f c = {};
#pragma unroll
        for (int kc = 0; kc < FIN / 4; ++kc) {
            const int k = kc * 4 + half * 2;
            v2f a, b;
            a[0] = arow[k + 0];
            a[1] = arow[k + 1];
            b[0] = W[(k + 0) * FOUT + nt * 16 + m];
            b[1] = W[(k + 1) * FOUT + nt * 16 + m];
            // 8 args: (neg_a, A, neg_b, B, c_mod, C, reuse_a, reuse_b)
            c = __builtin_amdgcn_wmma_f32_16x16x4_f32(false, a, false, b,
                                                      (short)0, c, false, false);
        }
#pragma unroll
        for (int r = 0; r < 8; ++r)
            Y[(size_t)(row0 + r + half * 8) * FOUT + nt * 16 + m] = c[r];
    }
}

// ---------------------------------------------------------------------------
// acc[i,f] = feat[i,f] * dinv[i]^2   (self-loop term; also initializes acc)
// ---------------------------------------------------------------------------
__global__ void selfloop_init_kernel(const float* __restrict__ feat,
                                     const float* __restrict__ dinv,
                                     float* __restrict__ acc,
                                     int n, int logF) {
    int i = blockIdx.x * blockDim.x + threadIdx.x;
    if (i < (n << logF)) {
        float di = dinv[i >> logF];
        acc[i] = feat[i] * di * di;
    }
}

// ---------------------------------------------------------------------------
// Edge scatter: acc[dst] += feat[src] * dinv[src]*dinv[dst]
// thread = (edge, 4-feature chunk); 2^shift chunks per edge (F = 4<<shift).
// Consecutive lanes cover one contiguous feature row -> coalesced gathers.
// ---------------------------------------------------------------------------
__global__ void scatter_kernel(const long long* __restrict__ src,
                               const long long* __restrict__ dst,
                               const float* __restrict__ dinv,
                               const float* __restrict__ feat,
                               float* __restrict__ acc,
                               int E, int shift) {
    int gid = blockIdx.x * blockDim.x + threadIdx.x;
    int e = gid >> shift;
    if (e >= E) return;
    int c = (gid & ((1 << shift) - 1)) * 4;
    int F = 4 << shift;
    int s = (int)src[e];
    int d = (int)dst[e];
    float nrm = dinv[s] * dinv[d];
    const float4 v = *(const float4*)(feat + (size_t)s * F + c);
    float* ap = acc + (size_t)d * F + c;
    atomicAdd(ap + 0, v.x * nrm);
    atomicAdd(ap + 1, v.y * nrm);
    atomicAdd(ap + 2, v.z * nrm);
    atomicAdd(ap + 3, v.w * nrm);
}

// ---------------------------------------------------------------------------
// a[i] = relu(a[i] + bias[i & (F-1)])   (in place)
// ---------------------------------------------------------------------------
__global__ void bias_relu_kernel(float* __restrict__ a,
                                 const float* __restrict__ bias,
                                 int total, int fmask) {
    int i = blockIdx.x * blockDim.x + threadIdx.x;
    if (i < total) {
        float v = a[i] + bias[i & fmask];
        a[i] = v > 0.0f ? v : 0.0f;
    }
}

// ---------------------------------------------------------------------------
extern "C" void kernel_launch(void* const* d_in, const int* in_sizes, int n_in,
                              void* d_out, int out_size, void* d_ws, size_t ws_size,
                              hipStream_t stream) {
    const float*     x  = (const float*)d_in[0];
    const long long* ei = (const long long*)d_in[1];   // int64 [2, E]
    const float*     W1 = (const float*)d_in[2];       // [16,32]
    const float*     b1 = (const float*)d_in[3];       // [32]
    const float*     W2 = (const float*)d_in[4];       // [32,64]
    const float*     b2 = (const float*)d_in[5];       // [64]
    float*           out = (float*)d_out;              // [N,64]

    const int N = in_sizes[0] / 16;
    const int E = in_sizes[1] / 2;
    const long long* src = ei;
    const long long* dst = ei + E;

    // Workspace layout (floats): dinv[N] | xw[N*32] | acc1[N*32] | hw[N*64]
    float* ws   = (float*)d_ws;
    size_t off  = 0;
    float* dinv = ws + off; off += ((size_t)N + 63) & ~(size_t)63;
    float* xw   = ws + off; off += (size_t)N * 32;
    float* acc1 = ws + off; off += (size_t)N * 32;
    float* hw   = ws + off; off += (size_t)N * 64;
    (void)ws_size;

    const int B = 256;
    auto blocks = [](long long work, int b) { return (int)((work + b - 1) / b); };

    // --- degree + normalization ---
    deg_init_kernel<<<blocks(N, B), B, 0, stream>>>(dinv, N);
    deg_count_kernel<<<blocks(E, B), B, 0, stream>>>(dst, dinv, E);
    deg_rsqrt_kernel<<<blocks(N, B), B, 0, stream>>>(dinv, N);

    // --- layer 1: xw = x @ W1 (WMMA f32), acc1 = self-loop + edge scatter ---
    wmma_gemm_kernel<16, 32><<<N / 16, 32, 0, stream>>>(x, W1, xw);
    selfloop_init_kernel<<<blocks((long long)N * 32, B), B, 0, stream>>>(xw, dinv, acc1, N, 5);
    scatter_kernel<<<blocks((long long)E * 8, B), B, 0, stream>>>(src, dst, dinv, xw, acc1, E, 3);
    bias_relu_kernel<<<blocks((long long)N * 32, B), B, 0, stream>>>(acc1, b1, N * 32, 31);

    // --- layer 2: hw = h @ W2 (WMMA f32), out = self-loop + edge scatter ---
    wmma_gemm_kernel<32, 64><<<N / 16, 32, 0, stream>>>(acc1, W2, hw);
    selfloop_init_kernel<<<blocks((long long)N * 64, B), B, 0, stream>>>(hw, dinv, out, N, 6);
    scatter_kernel<<<blocks((long long)E * 16, B), B, 0, stream>>>(src, dst, dinv, hw, out, E, 4);
    bias_relu_kernel<<<blocks((long long)N * 64, B), B, 0, stream>>>(out, b2, N * 64, 63);
}